// PoxelGCN_55886114456065
// MI455X (gfx1250) — compile-verified
//
#include <hip/hip_runtime.h>
#include <math.h>

// ---------------------------------------------------------------------------
// PoxelGCN forward for MI455X (gfx1250).
// Roofline: ~13 GFLOP fp32 GEMM + graph ops over ~100MB working set; the
// whole model is L2-resident (192MB), so HBM BW (23.3 TB/s) is not the limit.
// GEMMs use the native fp32 matrix op V_WMMA_F32_16X16X4_F32 (exact fp32).
// Register blocking: one wave32 owns a 16x64 C strip = 4 independent 16x16
// accumulators -> the A fragment is loaded once per k-step and the 4 WMMAs
// have no RAW dependency on each other, keeping the matrix pipe busy while
// the next k-step's loads are in flight.
// ---------------------------------------------------------------------------

#define HH    128
#define HEADS 3
#define NB    16     // batch of graphs

typedef __attribute__((ext_vector_type(2))) float v2f;
typedef __attribute__((ext_vector_type(8))) float v8f;

// order-preserving float<->uint map for atomicMax-based segment max
__device__ __forceinline__ unsigned fenc(float f) {
  unsigned u = __float_as_uint(f);
  return (u & 0x80000000u) ? ~u : (u | 0x80000000u);
}
__device__ __forceinline__ float fdec(unsigned u) {
  return (u & 0x80000000u) ? __uint_as_float(u & 0x7FFFFFFFu)
                           : __uint_as_float(~u);
}

// ---------------------------- BatchNorm ------------------------------------
__global__ void k_bn_stats(const float* __restrict__ x, float* __restrict__ stats,
                           int rows) {
  int c = threadIdx.x;                       // 128 threads = columns
  float s = 0.f, s2 = 0.f;
  for (int r = blockIdx.x; r < rows; r += gridDim.x) {
    float v = x[(size_t)r * HH + c];
    s += v; s2 += v * v;
  }
  atomicAdd(&stats[c], s);
  atomicAdd(&stats[HH + c], s2);
}

__global__ void k_bn_finalize(float* stats, int rows) {
  int c = threadIdx.x;
  float inv  = 1.f / (float)rows;
  float mean = stats[c] * inv;
  float var  = stats[HH + c] * inv - mean * mean;
  stats[c]      = mean;
  stats[HH + c] = rsqrtf(var + 1e-5f);
}

__global__ void k_bn_apply(const float* __restrict__ x, float* __restrict__ y,
                           const float* __restrict__ stats,
                           const float* __restrict__ g, const float* __restrict__ b,
                           int total) {
  int i = blockIdx.x * blockDim.x + threadIdx.x;
  if (i >= total) return;
  int c = i & (HH - 1);
  y[i] = (x[i] - stats[c]) * stats[HH + c] * g[c] + b[c];
}

// ------------------------- fp32 WMMA GEMM ----------------------------------
// Y[M,N] = X[M,K] @ W[K,N]; one wave per 16x64 strip (4 accumulators).
// N must be a multiple of 64, M a multiple of 16, K a multiple of 4.
__global__ __launch_bounds__(256) void k_gemm_wmma(
    const float* __restrict__ X, const float* __restrict__ W,
    float* __restrict__ Y, int M, int N, int K) {
  int lane  = threadIdx.x & 31;
  int wave  = threadIdx.x >> 5;
  int tileN = blockIdx.x;                    // N/64 column groups
  int tileM = blockIdx.y * 8 + wave;
  if (tileM * 16 >= M) return;               // wave-uniform (EXEC stays all-1s)
  int half = lane >> 4;                      // 0: K pair {k,k+1}, 1: {k+2,k+3}
  int r    = lane & 15;
  int arow = tileM * 16 + r;                 // A-matrix row for this lane
  int col0 = tileN * 64 + r;                 // first B/C column for this lane
  v8f acc0 = {0.f, 0.f, 0.f, 0.f, 0.f, 0.f, 0.f, 0.f};
  v8f acc1 = acc0, acc2 = acc0, acc3 = acc0;
  for (int k0 = 0; k0 < K; k0 += 4) {
    int ka = k0 + 2 * half;
    v2f a;
    a.x = X[(size_t)arow * K + ka];
    a.y = X[(size_t)arow * K + ka + 1];
    const float* w0 = W + (size_t)ka * N + col0;
    const float* w1 = W + (size_t)(ka + 1) * N + col0;
    v2f b0; b0.x = w0[0];  b0.y = w1[0];
    v2f b1; b1.x = w0[16]; b1.y = w1[16];
    v2f b2; b2.x = w0[32]; b2.y = w1[32];
    v2f b3; b3.x = w0[48]; b3.y = w1[48];
    acc0 = __builtin_amdgcn_wmma_f32_16x16x4_f32(false, a, false, b0,
                                                 (short)0, acc0, false, false);
    acc1 = __builtin_amdgcn_wmma_f32_16x16x4_f32(false, a, false, b1,
                                                 (short)0, acc1, false, false);
    acc2 = __builtin_amdgcn_wmma_f32_16x16x4_f32(false, a, false, b2,
                                                 (short)0, acc2, false, false);
    acc3 = __builtin_amdgcn_wmma_f32_16x16x4_f32(false, a, false, b3,
                                                 (short)0, acc3, false, false);
  }
  int base = tileM * 16 + half * 8;          // C layout: VGPR v -> row v + 8*half
#pragma unroll
  for (int v = 0; v < 8; ++v) {
    float* yrow = Y + (size_t)(base + v) * N + col0;
    yrow[0]  = acc0[v];
    yrow[16] = acc1[v];
    yrow[32] = acc2[v];
    yrow[48] = acc3[v];
  }
}

// --------------------------- top-k pooling ---------------------------------
// One block per graph. softmax(h@w) over n nodes, bitonic sort desc in LDS,
// keep top `keep`, write h*val and gathered positions.
__global__ __launch_bounds__(1024) void k_topk_pool(
    const float* __restrict__ h, const float* __restrict__ pos,
    const float* __restrict__ w, float* __restrict__ hout,
    float* __restrict__ posout, int n, int keep) {
  __shared__ float sc[2048];
  __shared__ int   si[2048];
  __shared__ float red[1024];
  int g = blockIdx.x, tid = threadIdx.x, nt = blockDim.x;

  for (int i = tid; i < n; i += nt) {
    const float* row = h + (size_t)(g * n + i) * HH;
    float acc = 0.f;
    for (int c = 0; c < HH; ++c) acc += row[c] * w[c];
    sc[i] = acc; si[i] = i;
  }
  __syncthreads();
  float lm = -1e30f;
  for (int i = tid; i < n; i += nt) lm = fmaxf(lm, sc[i]);
  red[tid] = lm; __syncthreads();
  for (int s = 512; s > 0; s >>= 1) { if (tid < s) red[tid] = fmaxf(red[tid], red[tid + s]); __syncthreads(); }
  float mx = red[0]; __syncthreads();
  float ls = 0.f;
  for (int i = tid; i < n; i += nt) { float e = expf(sc[i] - mx); sc[i] = e; ls += e; }
  red[tid] = ls; __syncthreads();
  for (int s = 512; s > 0; s >>= 1) { if (tid < s) red[tid] += red[tid + s]; __syncthreads(); }
  float inv = 1.f / red[0]; __syncthreads();
  for (int i = tid; i < n; i += nt) sc[i] *= inv;
  __syncthreads();

  for (int size = 2; size <= n; size <<= 1) {
    for (int stride = size >> 1; stride > 0; stride >>= 1) {
      for (int i = tid; i < n; i += nt) {
        int j = i ^ stride;
        if (j > i) {
          bool descBlk = ((i & size) == 0);
          float a = sc[i], b = sc[j];
          bool sw = descBlk ? (a < b) : (a > b);
          if (sw) {
            sc[i] = b; sc[j] = a;
            int t = si[i]; si[i] = si[j]; si[j] = t;
          }
        }
      }
      __syncthreads();
    }
  }

  for (int t = tid; t < keep * 3; t += nt) {
    int m = t / 3, cc = t - m * 3;
    posout[(size_t)(g * keep + m) * 3 + cc] = pos[(size_t)(g * n + si[m]) * 3 + cc];
  }
  for (int t = tid; t < keep * HH; t += nt) {
    int m = t >> 7, c = t & (HH - 1);
    hout[(size_t)(g * keep + m) * HH + c] =
        h[(size_t)(g * n + si[m]) * HH + c] * sc[m];
  }
}

// ----------------------- attention readout ---------------------------------
// r = (softmax(h@gw+gb)^T h) @ roW + roB  -> cat[:, coloff:coloff+256]
__global__ __launch_bounds__(256) void k_att_agg(
    const float* __restrict__ h, const float* __restrict__ gw,
    const float* __restrict__ gb, const float* __restrict__ roW,
    const float* __restrict__ roB, float* __restrict__ cat, int coloff, int n) {
  __shared__ float alpha[1024];
  __shared__ float red[256];
  __shared__ float hbar[HH];
  int g = blockIdx.x, tid = threadIdx.x;
  float gbv = gb[0];
  for (int i = tid; i < n; i += 256) {
    const float* row = h + (size_t)(g * n + i) * HH;
    float acc = gbv;
    for (int c = 0; c < HH; ++c) acc += row[c] * gw[c];
    alpha[i] = acc;
  }
  __syncthreads();
  float lm = -1e30f;
  for (int i = tid; i < n; i += 256) lm = fmaxf(lm, alpha[i]);
  red[tid] = lm; __syncthreads();
  for (int s = 128; s > 0; s >>= 1) { if (tid < s) red[tid] = fmaxf(red[tid], red[tid + s]); __syncthreads(); }
  float mx = red[0]; __syncthreads();
  float ls = 0.f;
  for (int i = tid; i < n; i += 256) { float e = expf(alpha[i] - mx); alpha[i] = e; ls += e; }
  red[tid] = ls; __syncthreads();
  for (int s = 128; s > 0; s >>= 1) { if (tid < s) red[tid] += red[tid + s]; __syncthreads(); }
  float inv = 1.f / red[0]; __syncthreads();
  if (tid < HH) {
    float acc = 0.f;
    for (int i = 0; i < n; ++i) acc += alpha[i] * h[(size_t)(g * n + i) * HH + tid];
    hbar[tid] = acc * inv;
  }
  __syncthreads();
  {
    int j = tid;                                   // 0..255
    float acc = roB[j];
    for (int c = 0; c < HH; ++c) acc += hbar[c] * roW[c * (2 * HH) + j];
    cat[(size_t)g * 768 + coloff + j] = acc;
  }
}

// ------------------------------- kNN ---------------------------------------
// thread-per-node top-10 smallest d2 (self included), LDS candidate tiles.
__global__ __launch_bounds__(256) void k_knn(const float* __restrict__ pos,
                                             int* __restrict__ knn, int n) {
  __shared__ float cx[256], cy[256], cz[256];
  int bpg = n >> 8;
  int g = blockIdx.x / bpg;
  int base = (blockIdx.x % bpg) << 8;
  int tid = threadIdx.x;
  int node = g * n + base + tid;
  float px = pos[node * 3], py = pos[node * 3 + 1], pz = pos[node * 3 + 2];
  float bd[10]; int bi[10];
#pragma unroll
  for (int j = 0; j < 10; ++j) { bd[j] = 3.0e38f; bi[j] = 0; }
  for (int c0 = 0; c0 < n; c0 += 256) {
    int cg = g * n + c0 + tid;
    cx[tid] = pos[cg * 3]; cy[tid] = pos[cg * 3 + 1]; cz[tid] = pos[cg * 3 + 2];
    __syncthreads();
    for (int t = 0; t < 256; ++t) {
      float dx = px - cx[t], dy = py - cy[t], dz = pz - cz[t];
      float nd = dx * dx + dy * dy + dz * dz;
      int ni = g * n + c0 + t;
      if (nd < bd[9]) {
#pragma unroll
        for (int j = 0; j < 10; ++j) {
          if (nd < bd[j]) { float td = bd[j]; int ti = bi[j];
                            bd[j] = nd; bi[j] = ni; nd = td; ni = ti; }
        }
      }
    }
    __syncthreads();
  }
#pragma unroll
  for (int j = 0; j < 10; ++j) knn[node * 10 + j] = bi[j];
}

// ---------------------------- edge MLP -------------------------------------
__device__ __forceinline__ float edge_mlp(float d, const float* w1,
                                          const float* b1, const float* w2,
                                          float b2) {
  float acc = b2;
#pragma unroll
  for (int j = 0; j < 32; ++j) acc += fmaxf(fmaf(d, w1[j], b1[j]), 0.f) * w2[j];
  return fmaxf(acc, 0.f);
}

__global__ void k_edge_w(const float* __restrict__ pos, const int* __restrict__ knn,
                         float* __restrict__ ew, const float* __restrict__ w1,
                         const float* __restrict__ b1, const float* __restrict__ w2,
                         const float* __restrict__ b2, int Nn) {
  int gid = blockIdx.x * blockDim.x + threadIdx.x;
  if (gid == 0) ew[(size_t)Nn * 10] = edge_mlp(0.f, w1, b1, w2, b2[0]); // loop edges
  if (gid >= Nn * 10) return;
  int i = gid / 10;
  int s = knn[gid];
  float dx = pos[i * 3] - pos[s * 3];
  float dy = pos[i * 3 + 1] - pos[s * 3 + 1];
  float dz = pos[i * 3 + 2] - pos[s * 3 + 2];
  float d2 = dx * dx + dy * dy + dz * dz;
  float d  = d2 > 0.f ? sqrtf(d2) : 0.f;
  ew[gid] = edge_mlp(d, w1, b1, w2, b2[0]);
}

// --------------------------- GATv2 passes ----------------------------------
// Edge e in [0, 21*Nn): node=e/21; t=e%21: t<10 fwd (dst=node), t<20 rev, t==20 loop.
__device__ __forceinline__ void edge_decode(int e, const int* knn, int* s, int* d,
                                            int* j10) {
  int node = e / 21, t = e - node * 21;
  if (t < 10)      { *s = knn[node * 10 + t];        *d = node; *j10 = node * 10 + t; }
  else if (t < 20) { *s = node; *d = knn[node * 10 + (t - 10)]; *j10 = node * 10 + (t - 10); }
  else             { *s = node; *d = node; *j10 = -1; }
}

__global__ __launch_bounds__(256) void k_gat_logits(
    const float* __restrict__ xl, const float* __restrict__ xr,
    const int* __restrict__ knn, const float* __restrict__ ew,
    const float* __restrict__ We, const float* __restrict__ att,
    float* __restrict__ logits, unsigned* __restrict__ mmax, int Nn) {
  int wave = blockIdx.x * 8 + (threadIdx.x >> 5);
  int lane = threadIdx.x & 31;
  int E = Nn * 21;
  if (wave >= E) return;
  int s, d, j10;
  edge_decode(wave, knn, &s, &d, &j10);
  float w = (j10 >= 0) ? ew[j10] : ew[(size_t)Nn * 10];
  int c = lane * 4;
#pragma unroll
  for (int h = 0; h < HEADS; ++h) {
    float lg = 0.f;
#pragma unroll
    for (int cc = 0; cc < 4; ++cc) {
      int ch = c + cc;
      float z = xl[(size_t)s * 384 + h * HH + ch] +
                xr[(size_t)d * 384 + h * HH + ch] + w * We[h * HH + ch];
      z = z > 0.f ? z : 0.2f * z;                    // leaky_relu(0.2)
      lg += z * att[h * HH + ch];
    }
#pragma unroll
    for (int off = 16; off >= 1; off >>= 1) lg += __shfl_xor(lg, off, 32);
    if (lane == 0) {
      logits[(size_t)wave * 3 + h] = lg;
      atomicMax(&mmax[d * 3 + h], fenc(lg));
    }
  }
}

__global__ void k_gat_expsum(float* __restrict__ logits,
                             const unsigned* __restrict__ mmax,
                             float* __restrict__ denom,
                             const int* __restrict__ knn, int Nn) {
  int gid = blockIdx.x * blockDim.x + threadIdx.x;
  int E = Nn * 21;
  if (gid >= E * 3) return;
  int e = gid / 3, h = gid - e * 3;
  int s, d, j10;
  edge_decode(e, knn, &s, &d, &j10);
  float m  = fdec(mmax[d * 3 + h]);
  float ex = expf(logits[gid] - m);
  logits[gid] = ex;                                  // reuse buffer for ex
  atomicAdd(&denom[d * 3 + h], ex);
}

__global__ __launch_bounds__(256) void k_gat_accum(
    const float* __restrict__ xl, const float* __restrict__ exv,
    const float* __restrict__ denom, const int* __restrict__ knn,
    float* __restrict__ accum, int Nn) {
  int wave = blockIdx.x * 8 + (threadIdx.x >> 5);
  int lane = threadIdx.x & 31;
  int E = Nn * 21;
  if (wave >= E) return;
  int s, d, j10;
  edge_decode(wave, knn, &s, &d, &j10);
  float a0 = exv[(size_t)wave * 3 + 0] / denom[d * 3 + 0];
  float a1 = exv[(size_t)wave * 3 + 1] / denom[d * 3 + 1];
  float a2 = exv[(size_t)wave * 3 + 2] / denom[d * 3 + 2];
  int c = lane * 4;
#pragma unroll
  for (int cc = 0; cc < 4; ++cc) {
    int ch = c + cc;
    float v = (a0 * xl[(size_t)s * 384 + ch] +
               a1 * xl[(size_t)s * 384 + HH + ch] +
               a2 * xl[(size_t)s * 384 + 2 * HH + ch]) * (1.f / 3.f);
    atomicAdd(&accum[(size_t)d * HH + ch], v);
  }
}

__global__ void k_gat_finalize(float* __restrict__ h, const float* __restrict__ accum,
                               const float* __restrict__ bias, int total) {
  int i = blockIdx.x * blockDim.x + threadIdx.x;
  if (i >= total) return;
  int c = i & (HH - 1);
  float t = h[i] + accum[i] + bias[c];               // h + (head-mean + bias)
  h[i] = t / (1.f + expf(-t));                       // silu
}

// ------------------------------- head --------------------------------------
__global__ __launch_bounds__(128) void k_head(const float* __restrict__ cat,
    const float* __restrict__ W1, const float* __restrict__ b1,
    const float* __restrict__ W2, const float* __restrict__ b2,
    float* __restrict__ out) {
  __shared__ float hid[HH];
  int g = blockIdx.x, j = threadIdx.x;
  float acc = b1[j];
  for (int c = 0; c < 768; ++c) acc += cat[(size_t)g * 768 + c] * W1[c * HH + j];
  hid[j] = fmaxf(acc, 0.f);
  __syncthreads();
  float o = b2[j];
  for (int c = 0; c < HH; ++c) o += hid[c] * W2[c * HH + j];
  out[(size_t)g * HH + j] = o;
}

// -------------------------- workspace layout (floats) ----------------------
constexpr size_t OFF_STATS = 0;                        // 256
constexpr size_t OFF_SM    = 256;                      // mmax(u32)+denom: 98304
constexpr size_t OFF_H0    = 98560;                    // 32768*128
constexpr size_t OFF_H1    = OFF_H0 + 4194304;         // 16384*128
constexpr size_t OFF_H2    = OFF_H1 + 2097152;         // 8192*128
constexpr size_t OFF_POS1  = OFF_H2 + 1048576;         // 16384*3
constexpr size_t OFF_POS2  = OFF_POS1 + 49152;         // 8192*3
constexpr size_t OFF_IDX   = OFF_POS2 + 24576;         // 16384*10 ints
constexpr size_t OFF_EW    = OFF_IDX + 163840;         // 16384*10+1 (+pad)
constexpr size_t OFF_XL    = OFF_EW + 163872;          // 16384*384
constexpr size_t OFF_XR    = OFF_XL + 6291456;         // 16384*384
constexpr size_t OFF_LOG   = OFF_XR + 6291456;         // 344064*3
constexpr size_t OFF_ACC   = OFF_LOG + 1032192;        // 16384*128
constexpr size_t OFF_CAT   = OFF_ACC + 2097152;        // 16*768
// total ~23.6M floats (~95 MB)

extern "C" void kernel_launch(void* const* d_in, const int* in_sizes, int n_in,
                              void* d_out, int out_size, void* d_ws, size_t ws_size,
                              hipStream_t stream) {
  const float* x     = (const float*)d_in[0];
  const float* pos   = (const float*)d_in[1];
  const float* bng   = (const float*)d_in[2];
  const float* bnb   = (const float*)d_in[3];
  const float* Wl    = (const float*)d_in[4];
  const float* Wr    = (const float*)d_in[5];
  const float* We    = (const float*)d_in[6];
  const float* attw  = (const float*)d_in[7];
  const float* cbias = (const float*)d_in[8];
  const float* poolw = (const float*)d_in[9];
  const float* gatew = (const float*)d_in[10];
  const float* gateb = (const float*)d_in[11];
  const float* roW   = (const float*)d_in[12];
  const float* roB   = (const float*)d_in[13];
  const float* emw1  = (const float*)d_in[14];
  const float* emb1  = (const float*)d_in[15];
  const float* emw2  = (const float*)d_in[16];
  const float* emb2  = (const float*)d_in[17];
  const float* ow1   = (const float*)d_in[18];
  const float* ob1   = (const float*)d_in[19];
  const float* ow2   = (const float*)d_in[20];
  const float* ob2   = (const float*)d_in[21];
  float* out = (float*)d_out;

  float*    ws    = (float*)d_ws;
  float*    stats = ws + OFF_STATS;
  unsigned* mmax  = (unsigned*)(ws + OFF_SM);
  float*    h0    = ws + OFF_H0;
  float*    h1    = ws + OFF_H1;
  float*    h2    = ws + OFF_H2;
  float*    pos1  = ws + OFF_POS1;
  float*    pos2  = ws + OFF_POS2;
  int*      idxb  = (int*)(ws + OFF_IDX);
  float*    ewb   = ws + OFF_EW;
  float*    xlb   = ws + OFF_XL;
  float*    xrb   = ws + OFF_XR;
  float*    logb  = ws + OFF_LOG;
  float*    accb  = ws + OFF_ACC;
  float*    catb  = ws + OFF_CAT;

  auto run_bn = [&](const float* in, float* outp, int rows, int stage) {
    hipMemsetAsync(stats, 0, 256 * sizeof(float), stream);
    k_bn_stats<<<256, 128, 0, stream>>>(in, stats, rows);
    k_bn_finalize<<<1, 128, 0, stream>>>(stats, rows);
    int total = rows * HH;
    k_bn_apply<<<(total + 255) / 256, 256, 0, stream>>>(
        in, outp, stats, bng + stage * HH, bnb + stage * HH, total);
  };

  auto run_gat = [&](float* hbuf, int Nn, int l) {
    dim3 ggrid(384 / 64, (Nn / 16 + 7) / 8);           // 16x64 strips per wave
    k_gemm_wmma<<<ggrid, 256, 0, stream>>>(hbuf, Wl + (size_t)l * HH * 384, xlb, Nn, 384, HH);
    k_gemm_wmma<<<ggrid, 256, 0, stream>>>(hbuf, Wr + (size_t)l * HH * 384, xrb, Nn, 384, HH);
    float* denomb = ((float*)mmax) + (size_t)Nn * 3;   // contiguous with mmax
    hipMemsetAsync(mmax, 0, (size_t)Nn * 3 * 2 * sizeof(float), stream);
    hipMemsetAsync(accb, 0, (size_t)Nn * HH * sizeof(float), stream);
    int E = Nn * 21;
    k_gat_logits<<<(E + 7) / 8, 256, 0, stream>>>(
        xlb, xrb, idxb, ewb, We + l * 384, attw + l * 384, logb, mmax, Nn);
    k_gat_expsum<<<(E * 3 + 255) / 256, 256, 0, stream>>>(logb, mmax, denomb, idxb, Nn);
    k_gat_accum<<<(E + 7) / 8, 256, 0, stream>>>(xlb, logb, denomb, idxb, accb, Nn);
    k_gat_finalize<<<(Nn * HH + 255) / 256, 256, 0, stream>>>(
        hbuf, accb, cbias + l * HH, Nn * HH);
  };

  // ---- stage 0: BN(x) -> h0 ; pool to N1=1024/graph -----------------------
  run_bn(x, h0, NB * 2048, 0);
  k_topk_pool<<<NB, 1024, 0, stream>>>(h0, pos, poolw + 0, h1, pos1, 2048, 1024);
  // r1 -> cat[:, 512:768]
  k_att_agg<<<NB, 256, 0, stream>>>(h1, gatew + 0, gateb + 0, roW + 0, roB + 0,
                                    catb, 512, 1024);
  // kNN + edge weights (stage 1)
  k_knn<<<NB * 4, 256, 0, stream>>>(pos1, idxb, 1024);
  k_edge_w<<<(16384 * 10 + 255) / 256, 256, 0, stream>>>(
      pos1, idxb, ewb, emw1 + 0, emb1 + 0, emw2 + 0, emb2 + 0, 16384);
  // conv 0 + BN1, conv 1 + BN2
  run_gat(h1, 16384, 0); run_bn(h1, h1, 16384, 1);
  run_gat(h1, 16384, 1); run_bn(h1, h1, 16384, 2);

  // ---- stage 1: pool to N2=512/graph --------------------------------------
  k_topk_pool<<<NB, 1024, 0, stream>>>(h1, pos1, poolw + 128, h2, pos2, 1024, 512);
  // r2 -> cat[:, 256:512]
  k_att_agg<<<NB, 256, 0, stream>>>(h2, gatew + 128, gateb + 1, roW + 32768,
                                    roB + 256, catb, 256, 512);
  k_knn<<<NB * 2, 256, 0, stream>>>(pos2, idxb, 512);
  k_edge_w<<<(8192 * 10 + 255) / 256, 256, 0, stream>>>(
      pos2, idxb, ewb, emw1 + 32, emb1 + 32, emw2 + 32, emb2 + 1, 8192);
  // conv 2 + BN3, conv 3 + BN4
  run_gat(h2, 8192, 2); run_bn(h2, h2, 8192, 3);
  run_gat(h2, 8192, 3); run_bn(h2, h2, 8192, 4);
  // r3 -> cat[:, 0:256]
  k_att_agg<<<NB, 256, 0, stream>>>(h2, gatew + 256, gateb + 2, roW + 65536,
                                    roB + 512, catb, 0, 512);

  // ---- head ---------------------------------------------------------------
  k_head<<<NB, 128, 0, stream>>>(catb, ow1, ob1, ow2, ob2, out);
}